// SpanBasedChunker_38431367365286
// MI455X (gfx1250) — compile-verified
//
#include <hip/hip_runtime.h>
#include <hip/hip_bf16.h>
#include <math.h>

// Problem dims (from reference setup_inputs)
#define BB   32
#define LL   1024
#define NSP  32768
#define NGD  64
#define NEGINF (-1e30f)

// ---------------------------------------------------------------------------
// CDNA5 async global->LDS path (gfx1250). Guarded so host pass / older
// toolchains fall back to synchronous reads straight from L2.
// ---------------------------------------------------------------------------
#if defined(__gfx1250__) && \
    __has_builtin(__builtin_amdgcn_global_load_async_to_lds_b128) && \
    __has_builtin(__builtin_amdgcn_s_wait_asynccnt)
#define USE_ASYNC 1
#else
#define USE_ASYNC 0
#endif

#if USE_ASYNC
typedef __attribute__((ext_vector_type(4))) int v4i_t;
// Builtin signature (from hipcc diagnostic): pointers to v4i, global side in
// addrspace(1), LDS side in addrspace(3). Cast through generic v4i* first.
#define GLOBAL_V4(p) ((__attribute__((address_space(1))) v4i_t*)((v4i_t*)(void*)(p)))
#define LDS_V4(p)    ((__attribute__((address_space(3))) v4i_t*)((v4i_t*)(void*)(p)))

__device__ __forceinline__ void async_row_b128(const float* g, float* lds) {
  __builtin_amdgcn_global_load_async_to_lds_b128(
      GLOBAL_V4(const_cast<float*>(g)), LDS_V4(lds),
      /*imm offset*/0, /*cpol*/0);
}
#endif

// ---------------------------------------------------------------------------
// Kernel 1: zero the span table (B*L*L f32 = 128 MB -> resident in 192MB L2)
// ---------------------------------------------------------------------------
__global__ void zero_T_kernel(float4* __restrict__ T4, size_t n4) {
  size_t i = (size_t)blockIdx.x * blockDim.x + threadIdx.x;
  size_t stride = (size_t)gridDim.x * blockDim.x;
  float4 z = {0.f, 0.f, 0.f, 0.f};
  for (; i < n4; i += stride) T4[i] = z;
}

// ---------------------------------------------------------------------------
// Kernel 2: scatter-add span scores. Layout T2[b][e][s] (row per end-index e)
// so the DP reads contiguous 4KB rows.
// ---------------------------------------------------------------------------
__global__ void scatter_kernel(const int* __restrict__ spans,
                               const float* __restrict__ scores,
                               float* __restrict__ T) {
  int idx = blockIdx.x * blockDim.x + threadIdx.x;
  if (idx >= BB * NSP) return;
  int b = idx / NSP;
  int s = spans[2 * idx + 0] & (LL - 1);  // inputs in [0, L); cheap clamp
  int e = spans[2 * idx + 1] & (LL - 1);
  atomicAdd(&T[((size_t)b * LL + e) * LL + s], scores[idx]);
}

// ---------------------------------------------------------------------------
// Kernel 3: semi-Markov CRF forward DP. One block per batch element.
// beta[0..L] lives in LDS; each of the 1024 sequential steps reads one 4KB
// row of T2, prefetched into double-buffered LDS with async B128 loads so
// L2 latency overlaps the logsumexp of the previous step.
//
// Per-step logsumexp is two-phase (max reduce, then sum-of-exp reduce) so the
// dependent-transcendental chain is short: max hops are cheap VALU ops and the
// 4 exps per thread are independent. Exactly 3 barriers per step.
// ---------------------------------------------------------------------------
__global__ __launch_bounds__(256) void dp_kernel(const float* __restrict__ T,
                                                 float* __restrict__ betaOut) {
  __shared__ float beta[LL + 1];
#if USE_ASYNC
  __shared__ float row[2][LL];
#endif
  __shared__ float red_m[8];
  __shared__ float red_s[8];

  const int b   = blockIdx.x;
  const int tid = threadIdx.x;
  const int wid = tid >> 5;
  const float* Tb = T + (size_t)b * LL * LL;

  for (int j = tid; j <= LL; j += 256) beta[j] = (j == 0) ? 0.f : NEGINF;

#if USE_ASYNC
  // Prefetch row e=0 (4 floats / thread, contiguous B128 per lane).
  async_row_b128(Tb + 4 * tid, &row[0][4 * tid]);
#endif
  __syncthreads();

  for (int i = 1; i <= LL; ++i) {
#if USE_ASYNC
    const int cur = (i - 1) & 1;
    const int nxt = i & 1;
    // Safe to overwrite row[nxt]: all reads of it finished before the
    // previous step's phase-1 barrier.
    if (i < LL) {
      async_row_b128(Tb + (size_t)i * LL + 4 * tid, &row[nxt][4 * tid]);
      __builtin_amdgcn_s_wait_asynccnt(1);  // oldest (current row) done; async loads retire in order
    } else {
      __builtin_amdgcn_s_wait_asynccnt(0);
    }
#endif
    __syncthreads();  // row[cur] visible from all waves; beta[i-1] visible

    // ---- Phase 1: candidates + max reduction --------------------------
    float c[4];
    float extra = NEGINF;      // beta[i-1] - S(i-1,i-1), handled by thread 0
    float lm = NEGINF;
#pragma unroll
    for (int k = 0; k < 4; ++k) {
      int s = 4 * tid + k;
#if USE_ASYNC
      float r = row[cur][s];
#else
      float r = Tb[(size_t)(i - 1) * LL + s];
#endif
      c[k] = (s < i) ? (beta[s] + r) : NEGINF;
      lm = fmaxf(lm, c[k]);
    }
    if (tid == 0) {
#if USE_ASYNC
      float diag = row[cur][i - 1];
#else
      float diag = Tb[(size_t)(i - 1) * LL + (i - 1)];
#endif
      extra = beta[i - 1] - diag;
      lm = fmaxf(lm, extra);
    }
#pragma unroll
    for (int off = 1; off < 32; off <<= 1)
      lm = fmaxf(lm, __shfl_xor(lm, off, 32));
    if ((tid & 31) == 0) red_m[wid] = lm;
    __syncthreads();  // also fences all row[cur] reads before next prefetch

    // Every thread recomputes the global max from 8 wave maxima (broadcast
    // LDS reads) -- no second shuffle stage, no extra barrier.
    float m = red_m[0];
#pragma unroll
    for (int w = 1; w < 8; ++w) m = fmaxf(m, red_m[w]);

    // ---- Phase 2: sum of exp(c - m) reduction -------------------------
    float sum = __expf(c[0] - m) + __expf(c[1] - m) +
                __expf(c[2] - m) + __expf(c[3] - m);
    if (tid == 0) sum += __expf(extra - m);
#pragma unroll
    for (int off = 1; off < 32; off <<= 1)
      sum += __shfl_xor(sum, off, 32);
    if ((tid & 31) == 0) red_s[wid] = sum;
    __syncthreads();
    if (tid == 0) {
      float t = red_s[0];
#pragma unroll
      for (int w = 1; w < 8; ++w) t += red_s[w];
      beta[i] = (t > 0.f) ? (m + __logf(t)) : NEGINF;
    }
    // beta[i] visibility guaranteed by next iteration's top barrier
  }

  __syncthreads();
  for (int j = tid; j <= LL; j += 256) betaOut[b * (LL + 1) + j] = beta[j];
}

// ---------------------------------------------------------------------------
// Kernel 4: numerator gather + final scalar = sum_b (log_num_b - log_den_b)
// ---------------------------------------------------------------------------
__global__ __launch_bounds__(256) void final_kernel(const float* __restrict__ T,
                                                    const float* __restrict__ betaOut,
                                                    const int* __restrict__ gold_spans,
                                                    const int* __restrict__ gold_labels,
                                                    const float* __restrict__ gold_mask,
                                                    const float* __restrict__ token_mask,
                                                    float* __restrict__ out) {
  __shared__ float red[8];
  int tid = threadIdx.x;
  float acc = 0.f;

  for (int idx = tid; idx < BB * NGD; idx += 256) {
    int b = idx / NGD;
    int s = gold_spans[2 * idx + 0] & (LL - 1);
    int e = gold_spans[2 * idx + 1] & (LL - 1);
    float v    = T[((size_t)b * LL + e) * LL + s];
    float sign = 2.f * (float)gold_labels[idx] - 1.f;
    acc += v * gold_mask[idx] * sign;
  }

  if (tid < BB) {
    float cnt = 0.f;
    for (int j = 0; j < LL; ++j) cnt += token_mask[tid * LL + j];
    int n = (int)(cnt + 0.5f);
    n = (n < 0) ? 0 : (n > LL ? LL : n);
    acc -= betaOut[tid * (LL + 1) + n];
  }

#pragma unroll
  for (int off = 1; off < 32; off <<= 1) acc += __shfl_xor(acc, off, 32);
  if ((tid & 31) == 0) red[tid >> 5] = acc;
  __syncthreads();
  if (tid == 0) {
    float t = 0.f;
#pragma unroll
    for (int w = 0; w < 8; ++w) t += red[w];
    out[0] = t;
  }
}

// ---------------------------------------------------------------------------
extern "C" void kernel_launch(void* const* d_in, const int* in_sizes, int n_in,
                              void* d_out, int out_size, void* d_ws, size_t ws_size,
                              hipStream_t stream) {
  (void)in_sizes; (void)n_in; (void)out_size;

  const int*   spans       = (const int*)d_in[0];
  const float* span_scores = (const float*)d_in[1];
  const int*   gold_spans  = (const int*)d_in[2];
  const int*   gold_labels = (const int*)d_in[3];
  /* d_in[4] span_mask: unused by the reference math */
  const float* gold_mask   = (const float*)d_in[5];
  const float* token_mask  = (const float*)d_in[6];

  // Workspace layout: T2[b][e][s] (128 MB, L2-resident), then beta table.
  float* T       = (float*)d_ws;
  float* betaOut = (float*)((char*)d_ws + (size_t)BB * LL * LL * sizeof(float));
  (void)ws_size;  // requires ~128.13 MB of scratch

  size_t n4 = (size_t)BB * LL * LL / 4;
  zero_T_kernel<<<4096, 256, 0, stream>>>((float4*)T, n4);
  scatter_kernel<<<(BB * NSP + 255) / 256, 256, 0, stream>>>(spans, span_scores, T);
  dp_kernel<<<BB, 256, 0, stream>>>(T, betaOut);
  final_kernel<<<1, 256, 0, stream>>>(T, betaOut, gold_spans, gold_labels,
                                      gold_mask, token_mask, (float*)d_out);
}